// DeepRMSAFeatureExtractor_79714593014208
// MI455X (gfx1250) — compile-verified
//
#include <hip/hip_runtime.h>
#include <hip/hip_bf16.h>
#include <math.h>

typedef __attribute__((ext_vector_type(2))) float v2f;
typedef __attribute__((ext_vector_type(8))) float v8f;

#define NN     128
#define MMM    256
#define KPATH  5
#define HID    128
#define EF     17
#define BATCH  16
#define IN_DIM 86721
#define COMB   33089
#define K4END  33088   // 4-aligned portion of K; element 33088 handled in reduce

// offsets inside one input row
#define OFF_SD   0
#define OFF_SLOT 256
#define OFF_SPEC 261
#define OFF_LF   321
#define OFF_BET  4673
#define OFF_ADJ  4801

// offsets inside one x row (concat order: Hm, source_dest, slots, c_band, l_band, alpha)
#define XO_HM   0
#define XO_SD   16384
#define XO_SLOT 16640
#define XO_CB   16645
#define XO_LB   16675
#define XO_AL   16705

#define SPLITK 64
#define CHUNK  520     // 4-aligned; 63*520 = 32760, last split = 328 (4-aligned)

// ---------------------------------------------------------------------------
// Kernel 1: per-batch feature prep. Exploits rank-1 structure of H_t and Hm.
// One block per batch element, 128 threads = 4 wave32s; softmax rows are
// handled wave-per-row with __shfl_xor butterfly reductions (wave32).
// ---------------------------------------------------------------------------
__global__ __launch_bounds__(128) void prep_kernel(
    const float* __restrict__ in, const float* __restrict__ WH,
    const float* __restrict__ WE, const float* __restrict__ a_attn,
    float* __restrict__ X)
{
    const int b = blockIdx.x;
    const int t = threadIdx.x;
    const float* row = in + (size_t)b * IN_DIM;
    float* xr = X + (size_t)b * COMB;

    __shared__ float s_lf[EF];
    __shared__ float s_ec[HID];
    __shared__ float s_bet[NN];
    __shared__ float s_red[128];
    __shared__ float s_c[4];   // c1, c2, s_e, ec0

    if (t < EF) s_lf[t] = 0.f;
    __syncthreads();
    // column sums of link_features (256 x 17)
    for (int idx = t; idx < MMM * EF; idx += 128)
        atomicAdd(&s_lf[idx % EF], row[OFF_LF + idx]);
    __syncthreads();

    // ec[h] = mean_lf @ WE ; betweenness into LDS
    {
        float acc = 0.f;
        #pragma unroll
        for (int e = 0; e < EF; ++e) acc += s_lf[e] * WE[e * HID + t];
        s_ec[t]  = acc * (1.0f / MMM);
        s_bet[t] = row[OFF_BET + t];
    }
    __syncthreads();

    // three 128-wide block reductions: c1 = WH.a1, c2 = WH.a2, s_e = ec.a3
    for (int which = 0; which < 3; ++which) {
        float v;
        if (which == 0)      v = WH[t] * a_attn[t];
        else if (which == 1) v = WH[t] * a_attn[HID + t];
        else                 v = s_ec[t] * a_attn[2 * HID + t];
        s_red[t] = v; __syncthreads();
        for (int off = 64; off >= 1; off >>= 1) {
            if (t < off) s_red[t] += s_red[t + off];
            __syncthreads();
        }
        if (t == 0) s_c[which] = s_red[0];
        __syncthreads();
    }
    if (t == 0) s_c[3] = s_ec[0];
    __syncthreads();

    const float c1 = s_c[0], c2 = s_c[1], se = s_c[2], ec0 = s_c[3];
    const int lane = t & 31;
    const int wv   = t >> 5;

    // attention rows: wave wv handles rows wv, wv+4, ...  (128 cols = 4/lane)
    for (int i = wv; i < NN; i += 4) {
        const float base = s_bet[i] * c1 + se;
        float vals[4], ex[4];
        float mx = -1.0e30f;
        #pragma unroll
        for (int u = 0; u < 4; ++u) {
            const int j = lane + 32 * u;
            const float adj = row[OFF_ADJ + i * NN + j];
            const float e   = base + s_bet[j] * c2;
            vals[u] = (adj > 0.f) ? tanhf(e) : -1.0e9f;
            mx = fmaxf(mx, vals[u]);
        }
        for (int off = 16; off >= 1; off >>= 1)
            mx = fmaxf(mx, __shfl_xor(mx, off, 32));
        float sum = 0.f, wsum = 0.f;
        #pragma unroll
        for (int u = 0; u < 4; ++u) {
            const int j = lane + 32 * u;
            ex[u] = __expf(vals[u] - mx);
            sum  += ex[u];
            wsum += ex[u] * s_bet[j];
        }
        for (int off = 16; off >= 1; off >>= 1) {
            sum  += __shfl_xor(sum,  off, 32);
            wsum += __shfl_xor(wsum, off, 32);
        }
        const float inv = 1.0f / sum;
        const float wi  = ec0 * wsum * inv;       // Hm row scale
        #pragma unroll
        for (int u = 0; u < 4; ++u) {
            const int j = lane + 32 * u;
            xr[XO_AL + i * NN  + j] = ex[u] * inv;     // alpha
            xr[XO_HM + i * HID + j] = wi * WH[j];      // Hm (rank-1)
        }
    }

    // small concat pieces
    for (int idx = t; idx < 2 * NN; idx += 128) xr[XO_SD + idx] = row[OFF_SD + idx];
    if (t < KPATH) xr[XO_SLOT + t] = row[OFF_SLOT + t];
    if (t < 30) {
        const int k = t / 6, r = t % 6;
        xr[XO_CB + t] = row[OFF_SPEC + k * 12 + r];       // spectrum[:,k,0,r]
        xr[XO_LB + t] = row[OFF_SPEC + k * 12 + 6 + r];   // spectrum[:,k,1,r]
    }
}

// ---------------------------------------------------------------------------
// Kernel 2: split-K WMMA GEMM  (16 x 33088) @ (33088 x 128) -> partials.
// grid = (8 N-tiles, 64 K-splits), one wave32 per workgroup.
// Branchless inner loop: 1x b64 A-load + 2x b32 B-loads + 1 WMMA per K-step
// of 4. V_WMMA_F32_16X16X4_F32 layouts:
//   A/B per-lane: M(or N) = lane&15, K = 2*(lane>>4) + v
//   D  per-lane:  N = lane&15, M = r + 8*(lane>>4)
// K element 33088 (COMB-1) is folded into the reduce kernel.
// ---------------------------------------------------------------------------
__global__ __launch_bounds__(32) void gemm_w0_kernel(
    const float* __restrict__ X, const float* __restrict__ W0,
    float* __restrict__ part)
{
    const int nt   = blockIdx.x;          // 0..7   (N tile)
    const int sp   = blockIdx.y;          // 0..63  (K split)
    const int lane = threadIdx.x;
    const int m    = lane & 15;
    const int kh   = (lane >> 4) << 1;
    const int k0   = sp * CHUNK;
    const int kend = (k0 + CHUNK < K4END) ? (k0 + CHUNK) : K4END;
    const int iters = (kend - k0) >> 2;

    const float* xp = X  + (size_t)m * COMB + k0 + kh;
    const float* wp = W0 + (size_t)(k0 + kh) * HID + nt * 16 + m;

    v8f acc = {};
    #pragma unroll 4
    for (int it = 0; it < iters; ++it) {
        // stream W0 ~128 rows ahead (speculative; dropped if invalid)
        __builtin_prefetch(wp + (size_t)128 * HID, 0, 1);
        v2f a, bv;
        a.x  = xp[0];
        a.y  = xp[1];          // contiguous -> global_load_b64
        bv.x = wp[0];
        bv.y = wp[HID];
        acc = __builtin_amdgcn_wmma_f32_16x16x4_f32(
                  false, a, false, bv, (short)0, acc, false, false);
        xp += 4;
        wp += 4 * HID;
    }

    float* p = part + (size_t)(sp * 8 + nt) * 256;
    const int mb = (lane >> 4) << 3;
    #pragma unroll
    for (int r = 0; r < 8; ++r)
        p[(mb + r) * 16 + m] = acc[r];
}

// ---------------------------------------------------------------------------
// Kernel 3: reduce SPLITK partials + tail K element + bias + ReLU -> x1
// ---------------------------------------------------------------------------
__global__ __launch_bounds__(256) void reduce_kernel(
    const float* __restrict__ part, const float* __restrict__ X,
    const float* __restrict__ W0, const float* __restrict__ b0,
    float* __restrict__ x1)
{
    const int t = blockIdx.x * 256 + threadIdx.x;   // 0..2047
    if (t >= BATCH * HID) return;
    const int m  = t >> 7;
    const int n  = t & 127;
    const int nt = n >> 4;
    const int nl = n & 15;
    float s = 0.f;
    #pragma unroll
    for (int sp = 0; sp < SPLITK; ++sp)
        s += part[(size_t)(sp * 8 + nt) * 256 + m * 16 + nl];
    // leftover K element (index 33088) not covered by the 4-aligned GEMM
    s += X[(size_t)m * COMB + K4END] * W0[(size_t)K4END * HID + n];
    s += b0[n];
    x1[m * HID + n] = fmaxf(s, 0.f);
}

// ---------------------------------------------------------------------------
// Kernel 4: four chained 128x128 ReLU layers, one block of 8 waves.
// Each wave owns one 16-col output tile; A streamed from LDS, WMMA f32.
// ---------------------------------------------------------------------------
__global__ __launch_bounds__(256) void mlp_kernel(
    const float* __restrict__ x1, const float* __restrict__ Wr,
    const float* __restrict__ br, float* __restrict__ out)
{
    __shared__ float xa[BATCH * HID];
    __shared__ float xb[BATCH * HID];
    const int t    = threadIdx.x;
    const int lane = t & 31;
    const int nt   = t >> 5;          // wave id == N tile 0..7
    const int m    = lane & 15;
    const int kh   = (lane >> 4) << 1;

    for (int i = t; i < BATCH * HID; i += 256) xa[i] = x1[i];
    __syncthreads();

    float* cur = xa;
    float* nxt = xb;
    for (int l = 0; l < 4; ++l) {
        const float* W = Wr + (size_t)l * HID * HID;
        v8f acc = {};
        #pragma unroll 4
        for (int kk = 0; kk < HID; kk += 4) {
            v2f a, bv;
            a.x  = cur[m * HID + kk + kh];
            a.y  = cur[m * HID + kk + kh + 1];
            bv.x = W[(kk + kh)     * HID + nt * 16 + m];
            bv.y = W[(kk + kh + 1) * HID + nt * 16 + m];
            acc = __builtin_amdgcn_wmma_f32_16x16x4_f32(
                      false, a, false, bv, (short)0, acc, false, false);
        }
        const float bias = br[l * HID + nt * 16 + m];
        const int mb = (lane >> 4) << 3;
        __syncthreads();   // everyone done reading cur before nxt overwritten
        #pragma unroll
        for (int r = 0; r < 8; ++r)
            nxt[(mb + r) * HID + nt * 16 + m] = fmaxf(acc[r] + bias, 0.f);
        __syncthreads();
        float* tmp = cur; cur = nxt; nxt = tmp;
    }

    for (int i = t; i < BATCH * HID; i += 256) out[i] = cur[i];
}

// ---------------------------------------------------------------------------
// Workspace layout (floats):
//   X        : 16*33089      = 529424
//   partials : 64*8*256      = 131072
//   x1       : 16*128        =   2048
// total ~2.65 MB
// ---------------------------------------------------------------------------
extern "C" void kernel_launch(void* const* d_in, const int* in_sizes, int n_in,
                              void* d_out, int out_size, void* d_ws, size_t ws_size,
                              hipStream_t stream) {
    const float* in    = (const float*)d_in[0];
    const float* WH    = (const float*)d_in[1];
    const float* WE    = (const float*)d_in[2];
    const float* a_att = (const float*)d_in[3];
    const float* W0    = (const float*)d_in[4];
    const float* b0    = (const float*)d_in[5];
    const float* Wr    = (const float*)d_in[6];
    const float* br    = (const float*)d_in[7];
    float* out = (float*)d_out;

    float* ws   = (float*)d_ws;
    float* X    = ws;
    float* part = ws + (size_t)BATCH * COMB;
    float* x1   = part + (size_t)SPLITK * 8 * 256;

    prep_kernel<<<BATCH, 128, 0, stream>>>(in, WH, WE, a_att, X);
    gemm_w0_kernel<<<dim3(8, SPLITK), 32, 0, stream>>>(X, W0, part);
    reduce_kernel<<<(BATCH * HID + 255) / 256, 256, 0, stream>>>(part, X, W0, b0, x1);
    mlp_kernel<<<1, 256, 0, stream>>>(x1, Wr, br, out);
}